// TemporalGCN_56281251447362
// MI455X (gfx1250) — compile-verified
//
#include <hip/hip_runtime.h>
#include <stdint.h>

// Problem constants (match reference)
#define BB 4
#define NN 20000
#define EE 320000
#define FF 64
#define HH 128
#define LL 3
#define ROWS (BB * NN)            // 80000
#define BN_EPS 1e-5f

typedef __attribute__((ext_vector_type(2))) float v2f;
typedef __attribute__((ext_vector_type(8))) float v8f;

// ---------------------------------------------------------------------------
// WMMA f32 GEMM:  C[M x 128] = A[M x K] * W[K x 128] (+ bias)
// Block = 256 threads (8 waves). Block computes a 16-row x 128-col stripe.
// A tile (16 x K) and W (staged 64 rows at a time) are loaded into LDS with
// coalesced b128 loads; WMMA fragments are then fed from LDS (padded strides
// keep the b64 A-reads and b32 B-reads bank-conflict free across lane halves).
//
// A 16x4 f32 layout: lanes 0-15 M=0..15; VGPR0 = K{koff}, VGPR1 = K{koff+1},
// koff = 0 for lanes 0-15, 2 for lanes 16-31.
// C/D 16x16 f32 layout: VGPR r -> M = r (+8 for upper lanes), N = lane&15.
// ---------------------------------------------------------------------------
template <int K>
__global__ void gcn_wmma_gemm(const float* __restrict__ A,
                              const float* __restrict__ W,
                              const float* __restrict__ bias,
                              float* __restrict__ C) {
  constexpr int APAD = K + 4;        // A row stride (dwords): conflict-free b64
  constexpr int WROW = 64;           // W rows per stage (keeps LDS < 64KB)
  constexpr int WPAD = HH + 8;       // 136: conflict-free b32 across halves
  __shared__ float lA[16 * APAD];
  __shared__ float lW[WROW * WPAD];

  const int t    = threadIdx.x;
  const int wave = t >> 5;
  const int lane = t & 31;
  const int r16  = lane & 15;
  const int koff = (lane < 16) ? 0 : 2;
  const int tm   = blockIdx.x;                // 16-row tile index
  const int bcol = wave * 16 + r16;           // output column (wave = col tile)

  // Stage A tile (16 x K), fully coalesced float4
#pragma unroll
  for (int i = t * 4; i < 16 * K; i += 256 * 4) {
    const int row = i / K, col = i % K;
    *(float4*)&lA[row * APAD + col] =
        *(const float4*)&A[(size_t)(tm * 16 + row) * K + col];
  }

  v8f acc = {};
  for (int kk = 0; kk < K; kk += WROW) {
    __syncthreads();                          // lA ready / lW reusable
    // Stage W rows [kk, kk+WROW), fully coalesced float4
#pragma unroll
    for (int i = t * 4; i < WROW * HH; i += 256 * 4) {
      const int row = i / HH, col = i % HH;
      *(float4*)&lW[row * WPAD + col] =
          *(const float4*)&W[(size_t)(kk + row) * HH + col];
    }
    __syncthreads();
#pragma unroll
    for (int k0 = 0; k0 < WROW; k0 += 4) {
      v2f a = *(const v2f*)&lA[r16 * APAD + kk + k0 + koff];
      v2f b;
      b.x = lW[(k0 + koff) * WPAD + bcol];
      b.y = lW[(k0 + koff + 1) * WPAD + bcol];
      acc = __builtin_amdgcn_wmma_f32_16x16x4_f32(false, a, false, b,
                                                  (short)0, acc, false, false);
    }
  }

  const float bb = bias ? bias[bcol] : 0.0f;
  const int rbase = tm * 16 + ((lane < 16) ? 0 : 8);
#pragma unroll
  for (int r = 0; r < 8; ++r) {
    C[(size_t)(rbase + r) * HH + bcol] = acc[r] + bb;
  }
}

// ---------------------------------------------------------------------------
// Degree / normalization
// ---------------------------------------------------------------------------
__global__ void zero_u32(unsigned* p, int n) {
  int i = blockIdx.x * blockDim.x + threadIdx.x;
  if (i < n) p[i] = 0u;
}

__global__ void degree_count(const long long* __restrict__ dst, unsigned* deg, int e) {
  int i = blockIdx.x * blockDim.x + threadIdx.x;
  if (i < e) atomicAdd(&deg[(int)dst[i]], 1u);
}

__global__ void make_dinv(const unsigned* __restrict__ deg, float* __restrict__ dinv, int n) {
  int i = blockIdx.x * blockDim.x + threadIdx.x;
  if (i < n) dinv[i] = rsqrtf((float)(deg[i] + 1u));   // +1 self loop, always > 0
}

// ---------------------------------------------------------------------------
// x[b][n][c] = bias[c]  (accumulator init for scatter)
// ---------------------------------------------------------------------------
__global__ void init_bias(float* __restrict__ x, const float* __restrict__ b, int total) {
  int i = blockIdx.x * blockDim.x + threadIdx.x;
  if (i < total) x[i] = b[i & (HH - 1)];
}

// ---------------------------------------------------------------------------
// Message scatter: one wave per (batch, edge); lane handles 4 channels.
// Edge index forced into an SGPR via readfirstlane so the per-edge index /
// norm loads scalarize. Edges e in [E, E+N) are the self loops.
// ---------------------------------------------------------------------------
__global__ void scatter_messages(const float* __restrict__ h,
                                 const long long* __restrict__ srcIdx,
                                 const long long* __restrict__ dstIdx,
                                 const float* __restrict__ dinv,
                                 float* __restrict__ out) {
  const int gw0  = (blockIdx.x * blockDim.x + threadIdx.x) >> 5;
  const int lane = threadIdx.x & 31;
  const int EN   = EE + NN;
  if (gw0 >= EN * BB) return;
  const int gw = __builtin_amdgcn_readfirstlane(gw0);   // wave-uniform
  const int b  = gw / EN;
  const int e  = gw - b * EN;

  int s, d; float nrm;
  if (e < EE) {
    s = (int)srcIdx[e];
    d = (int)dstIdx[e];
    nrm = dinv[s] * dinv[d];
  } else {
    s = d = e - EE;
    const float dv = dinv[s];
    nrm = dv * dv;
  }
  const float4 hv = *(const float4*)(h + ((size_t)b * NN + s) * HH + lane * 4);
  float* op = out + ((size_t)b * NN + d) * HH + lane * 4;
  atomicAdd(op + 0, hv.x * nrm);
  atomicAdd(op + 1, hv.y * nrm);
  atomicAdd(op + 2, hv.z * nrm);
  atomicAdd(op + 3, hv.w * nrm);
}

// ---------------------------------------------------------------------------
// BatchNorm (two pass) + ReLU
// ---------------------------------------------------------------------------
__global__ void zero_f32(float* p, int n) {
  int i = blockIdx.x * blockDim.x + threadIdx.x;
  if (i < n) p[i] = 0.0f;
}

__global__ void bn_stats(const float* __restrict__ x, float* __restrict__ stats) {
  const int c = threadIdx.x & (HH - 1);
  const int slot   = (blockIdx.x * blockDim.x + threadIdx.x) >> 7;
  const int nslots = (gridDim.x * blockDim.x) >> 7;
  float s = 0.0f, s2 = 0.0f;
  for (int r = slot; r < ROWS; r += nslots) {
    const float v = x[(size_t)r * HH + c];
    s += v;
    s2 += v * v;
  }
  atomicAdd(&stats[c], s);
  atomicAdd(&stats[HH + c], s2);
}

__global__ void bn_relu(float* __restrict__ x, const float* __restrict__ stats,
                        const float* __restrict__ gamma, const float* __restrict__ beta,
                        int total) {
  int i = blockIdx.x * blockDim.x + threadIdx.x;
  if (i >= total) return;
  const int c = i & (HH - 1);
  const float invM = 1.0f / (float)ROWS;
  const float mean = stats[c] * invM;
  const float var  = stats[HH + c] * invM - mean * mean;
  float v = (x[i] - mean) * rsqrtf(var + BN_EPS) * gamma[c] + beta[c];
  x[i] = v > 0.0f ? v : 0.0f;
}

// ---------------------------------------------------------------------------
// Prediction head: one wave per row, float4 per lane, shuffle reduce.
// ---------------------------------------------------------------------------
__global__ void pred_head(const float* __restrict__ x, const float* __restrict__ predW,
                          const float* __restrict__ predb, float* __restrict__ out) {
  const int gw   = (blockIdx.x * blockDim.x + threadIdx.x) >> 5;
  const int lane = threadIdx.x & 31;
  if (gw >= ROWS) return;
  const float4 xv = *(const float4*)(x + (size_t)gw * HH + lane * 4);
  const float4 wv = *(const float4*)(predW + lane * 4);
  float s = xv.x * wv.x + xv.y * wv.y + xv.z * wv.z + xv.w * wv.w;
#pragma unroll
  for (int off = 16; off > 0; off >>= 1) s += __shfl_down(s, off, 32);
  if (lane == 0) out[gw] = s + predb[0];
}

// ---------------------------------------------------------------------------
extern "C" void kernel_launch(void* const* d_in, const int* in_sizes, int n_in,
                              void* d_out, int out_size, void* d_ws, size_t ws_size,
                              hipStream_t stream) {
  const float*     x_seq = (const float*)d_in[0];
  const long long* eidx  = (const long long*)d_in[1];     // int64 [2, E]
  const float*     embW  = (const float*)d_in[2];
  const float*     embb  = (const float*)d_in[3];
  const float*     convW = (const float*)d_in[4];          // [L,H,H]
  const float*     convb = (const float*)d_in[5];          // [L,H]
  const float*     gamma = (const float*)d_in[6];
  const float*     beta  = (const float*)d_in[7];
  const float*     predW = (const float*)d_in[8];
  const float*     predb = (const float*)d_in[9];
  float*           out   = (float*)d_out;

  const long long* srcI = eidx;        // edge_index[0]
  const long long* dstI = eidx + EE;   // edge_index[1]

  // Workspace layout
  float*    xbuf  = (float*)d_ws;                       // ROWS*HH
  float*    hbuf  = xbuf + (size_t)ROWS * HH;           // ROWS*HH
  unsigned* deg   = (unsigned*)(hbuf + (size_t)ROWS * HH);
  float*    dinv  = (float*)(deg + NN);
  float*    stats = dinv + NN;                          // 2*HH floats

  const int TPB = 256;
  const int total    = ROWS * HH;                       // 10,240,000
  const int elemBlks = (total + TPB - 1) / TPB;
  const int gemmBlks = ROWS / 16;                       // 5000 (M divisible by 16)
  const int scatBlks = ((EE + NN) * BB * 32 + TPB - 1) / TPB;
  const int predBlks = (ROWS * 32 + TPB - 1) / TPB;

  // 1) degree + D^-1/2
  zero_u32<<<(NN + TPB - 1) / TPB, TPB, 0, stream>>>(deg, NN);
  degree_count<<<(EE + TPB - 1) / TPB, TPB, 0, stream>>>(dstI, deg, EE);
  make_dinv<<<(NN + TPB - 1) / TPB, TPB, 0, stream>>>(deg, dinv, NN);

  // 2) embedding: x = x_seq @ embW + embb     [ROWS,64]x[64,128]
  gcn_wmma_gemm<FF><<<gemmBlks, TPB, 0, stream>>>(x_seq, embW, embb, xbuf);

  // 3) GCN layers
  for (int l = 0; l < LL; ++l) {
    // h = x @ convW[l]
    gcn_wmma_gemm<HH><<<gemmBlks, TPB, 0, stream>>>(
        xbuf, convW + (size_t)l * HH * HH, nullptr, hbuf);
    // x = convb[l] (accumulator init), then scatter normalized messages
    init_bias<<<elemBlks, TPB, 0, stream>>>(xbuf, convb + (size_t)l * HH, total);
    scatter_messages<<<scatBlks, TPB, 0, stream>>>(hbuf, srcI, dstI, dinv, xbuf);
    if (l < LL - 1) {
      zero_f32<<<1, 2 * HH, 0, stream>>>(stats, 2 * HH);
      bn_stats<<<512, TPB, 0, stream>>>(xbuf, stats);
      bn_relu<<<elemBlks, TPB, 0, stream>>>(xbuf, stats, gamma, beta, total);
    }
  }

  // 4) prediction head -> [B, N]
  pred_head<<<predBlks, TPB, 0, stream>>>(xbuf, predW, predb, out);
}